// DigitCaps_39230231281960
// MI455X (gfx1250) — compile-verified
//
#include <hip/hip_runtime.h>

typedef float v2f __attribute__((ext_vector_type(2)));
typedef float v8f __attribute__((ext_vector_type(8)));
typedef unsigned int u32x4 __attribute__((ext_vector_type(4)));
typedef int i32x4 __attribute__((ext_vector_type(4)));
typedef int i32x8 __attribute__((ext_vector_type(8)));

#define EPSF   1e-7f
#define Bsz    64
#define Ncap   2048
#define ILEN   8
#define Ccap   32
#define Llen   16
#define CL     512      // Ccap * Llen
#define JCHUNK 32
#define NWAVE  4
#define CPW    (Ccap / NWAVE)     // 8 capsules per wave
#define NCHUNK (Ncap / JCHUNK)    // 64 j-chunks
#define NBT    4                  // 64 batch / 16-row WMMA tiles
#define WBYTES (ILEN * CL * 4)    // 16384 bytes = one W[j] slice

// Issue a TDM 1-D copy of W[j] (16 KB) into LDS. Descriptor per ISA ch.8:
// group0 = {count=1, lds_addr, global_addr, type=2}, group1 = dims/strides.
__device__ __forceinline__ void tdm_load_wj(const float* Wm, int j,
                                            unsigned int lds_addr) {
  unsigned long long ga = (unsigned long long)(size_t)Wm +
                          (unsigned long long)j * WBYTES;
  u32x4 g0;
  g0[0] = 1u;                                  // count=1, user mode
  g0[1] = lds_addr;                            // LDS byte address
  g0[2] = (unsigned int)ga;                    // global_addr[31:0]
  g0[3] = (unsigned int)(ga >> 32) | 0x80000000u;  // addr[56:32] | type=2
  i32x8 g1;
  g1[0] = 0x20000;                // data_size=2 (4B elements)
  g1[1] = (int)(4096u << 16);     // tensor_dim0[15:0]=4096 at bits[63:48]
  g1[2] = (int)(1u << 16);        // tensor_dim0 hi=0 | tensor_dim1=1
  g1[3] = (int)(4096u << 16);     // tile_dim0=4096 at bits[127:112]
  g1[4] = 1;                      // tile_dim1=1, tile_dim2=0
  g1[5] = 4096;                   // tensor_dim0_stride lo
  g1[6] = 0;
  g1[7] = 0;
  i32x4 gz4 = {0, 0, 0, 0};
  i32x8 gz8 = {0, 0, 0, 0, 0, 0, 0, 0};
  __builtin_amdgcn_tensor_load_to_lds(g0, g1, gz4, gz4, gz8, 0);
}

// ---------------------------------------------------------------------------
// Routing pass. Workgroup = (16-batch tile) x (32-j chunk); 4 waves cooperate
// on the same j, each owning 8 capsules. u_hat tiles via V_WMMA_F32_16X16X4_F32
// (A = u rows over batch, B = W[j] slice from LDS; each 16-wide N tile is one
// capsule's L=16 lanes). W[j] staged by the Tensor Data Mover, double-buffered.
// ---------------------------------------------------------------------------
__global__ __launch_bounds__(128) void caps_route(
    const float* __restrict__ x,      // [B, N, iL]
    const float* __restrict__ Wm,     // [N, iL, CL]
    const float* __restrict__ vsum,   // [B, CL] running sum of v's
    float* __restrict__ s_part,      // [NCHUNK, B, CL] partials
    int routed)
{
  __shared__ float s_lds[16 * CL];          // 32 KB s accumulator (b-tile)
  __shared__ float v_lds[16 * CL];          // 32 KB vsum tile
  __shared__ float w_lds[2][ILEN * CL];     // 32 KB W[j] double buffer (TDM)
  __shared__ float agr[16][Ccap + 1];       // logits -> softmax coeffs

  const int tid  = threadIdx.x;
  const int wave = tid >> 5;
  const int lane = tid & 31;
  const int lh   = lane >> 4;               // lane half
  const int col  = lane & 15;
  const int jchunk = blockIdx.x;
  const int btile  = blockIdx.y;
  const int b0 = btile * 16;
  const int jbase = jchunk * JCHUNK;

  // prologue: wave0 kicks the TDM for j=0 while everyone initializes LDS
  if (wave == 0) {
    tdm_load_wj(Wm, jbase, (unsigned int)(size_t)(&w_lds[0][0]));
  }
  for (int i = tid; i < 16 * CL; i += 128) {
    s_lds[i] = 0.0f;
    v_lds[i] = routed ? vsum[(size_t)(b0 + (i >> 9)) * CL + (i & 511)] : 0.0f;
  }

  for (int jj = 0; jj < JCHUNK; ++jj) {
    const int j = jbase + jj;
    if (wave == 0) {
      if (jj + 1 < JCHUNK) {   // prefetch next W slice, wait for current
        tdm_load_wj(Wm, j + 1,
                    (unsigned int)(size_t)(&w_lds[(jj + 1) & 1][0]));
        __builtin_amdgcn_s_wait_tensorcnt(1);
      } else {
        __builtin_amdgcn_s_wait_tensorcnt(0);
      }
    }
    __syncthreads();                         // buffer jj visible to all waves
    const float* wj = &w_lds[jj & 1][0];

    // A operand (16x4 f32): lane -> M=col, VGPR p -> K = 2*lh + p (+4 step 1)
    const float* ub = x + ((size_t)(b0 + col) * Ncap + j) * ILEN + 2 * lh;
    v2f a0; a0.x = ub[0]; a0.y = ub[1];
    v2f a1; a1.x = ub[4]; a1.y = ub[5];

    // u_hat tiles for this wave's 8 capsules, kept in registers (64 VGPRs)
    v8f accs[CPW];
#pragma unroll
    for (int ci = 0; ci < CPW; ++ci) {
      const int o = (wave * CPW + ci) * 16 + col;
      v2f bm0, bm1;
      bm0.x = wj[(2 * lh + 0) * CL + o];
      bm0.y = wj[(2 * lh + 1) * CL + o];
      bm1.x = wj[(2 * lh + 4) * CL + o];
      bm1.y = wj[(2 * lh + 5) * CL + o];
      v8f acc = {};
      acc = __builtin_amdgcn_wmma_f32_16x16x4_f32(false, a0, false, bm0,
                                                  (short)0, acc, false, false);
      acc = __builtin_amdgcn_wmma_f32_16x16x4_f32(false, a1, false, bm1,
                                                  (short)0, acc, false, false);
      accs[ci] = acc;
    }

    if (routed) {
      // agreement logits b[b,j,c] = u_hat . vsum ; 8 independent shuffle
      // chains interleaved per xor step so the ds waits amortize.
#pragma unroll
      for (int ci = 0; ci < CPW; ++ci) {
        const int c = wave * CPW + ci;
        float t[8];
#pragma unroll
        for (int k = 0; k < 8; ++k)
          t[k] = accs[ci][k] * v_lds[(k + 8 * lh) * CL + c * 16 + col];
#pragma unroll
        for (int off = 1; off <= 8; off <<= 1) {
#pragma unroll
          for (int k = 0; k < 8; ++k)
            t[k] += __shfl_xor(t[k], off, 32);
        }
        if (col == 0) {
#pragma unroll
          for (int k = 0; k < 8; ++k) agr[k + 8 * lh][c] = t[k];
        }
      }
      __syncthreads();
      // softmax over capsules, one row per lane of wave 0
      if (wave == 0 && lane < 16) {
        float mx = -3.4e38f;
        for (int c = 0; c < Ccap; ++c) mx = fmaxf(mx, agr[lane][c]);
        float ssum = 0.0f;
        for (int c = 0; c < Ccap; ++c) {
          float e = __expf(agr[lane][c] - mx);
          agr[lane][c] = e;
          ssum += e;
        }
        float inv = 1.0f / ssum;
        for (int c = 0; c < Ccap; ++c) agr[lane][c] *= inv;
      }
      __syncthreads();
    }

    // s += coeff * u_hat ; capsule ranges are wave-disjoint -> plain LDS rmw
#pragma unroll
    for (int ci = 0; ci < CPW; ++ci) {
      const int c = wave * CPW + ci;
#pragma unroll
      for (int k = 0; k < 8; ++k) {
        const int row = k + 8 * lh;
        const float coef = routed ? agr[row][c] : (1.0f / 32.0f);
        s_lds[row * CL + c * 16 + col] += coef * accs[ci][k];
      }
    }
    __syncthreads();   // protect agr/w_lds reuse next iteration
  }

  // deterministic partial per workgroup (no global float atomics)
  for (int i = tid; i < 16 * CL; i += 128) {
    s_part[((size_t)jchunk * Bsz + b0 + (i >> 9)) * CL + (i & 511)] = s_lds[i];
  }
}

// ---------------------------------------------------------------------------
// Reduce partials, add bias, squash; accumulate vsum or write final output.
// ---------------------------------------------------------------------------
__global__ __launch_bounds__(32) void caps_squash(
    const float* __restrict__ s_part,
    const float* __restrict__ bias,
    float* __restrict__ vsum,
    float* __restrict__ out,
    int final_)
{
  const int bc = blockIdx.x;
  const int b = bc >> 5;
  const int c = bc & 31;
  const int l = threadIdx.x & 15;

  float s = 0.0f;
  for (int p = 0; p < NCHUNK; ++p)
    s += s_part[((size_t)p * Bsz + b) * CL + c * 16 + l];
  s += bias[c * 16 + l];

  float t = s * s;
  t += __shfl_xor(t, 1, 32);
  t += __shfl_xor(t, 2, 32);
  t += __shfl_xor(t, 4, 32);
  t += __shfl_xor(t, 8, 32);
  const float n2 = t + EPSF;
  const float v = n2 / (1.0f + n2) * rsqrtf(n2) * s;

  if (threadIdx.x < 16) {
    const size_t idx = (size_t)b * CL + c * 16 + l;
    if (final_) out[idx] = v;
    else        vsum[idx] += v;
  }
}

__global__ void caps_zero(float* __restrict__ p, int n) {
  int i = blockIdx.x * blockDim.x + threadIdx.x;
  if (i < n) p[i] = 0.0f;
}

extern "C" void kernel_launch(void* const* d_in, const int* in_sizes, int n_in,
                              void* d_out, int out_size, void* d_ws, size_t ws_size,
                              hipStream_t stream) {
  const float* x    = (const float*)d_in[0];   // [64,8,8,32,8] -> [64,2048,8]
  const float* Wm   = (const float*)d_in[1];   // [2048,8,512]
  const float* bias = (const float*)d_in[2];   // [32,16]
  float* out = (float*)d_out;                  // [64,32,16]

  // ws: s_part = NCHUNK*B*CL floats (8 MB), vsum = B*CL floats (128 KB)
  float* s_part = (float*)d_ws;
  float* vsum   = s_part + (size_t)NCHUNK * Bsz * CL;

  caps_zero<<<(Bsz * CL + 255) / 256, 256, 0, stream>>>(vsum, Bsz * CL);

  dim3 grid(NCHUNK, NBT);
  // r = 0: uniform coefficients (softmax of zeros) = 1/32
  caps_route<<<grid, 128, 0, stream>>>(x, Wm, vsum, s_part, 0);
  caps_squash<<<Bsz * Ccap, 32, 0, stream>>>(s_part, bias, vsum, out, 0); // vsum = v0
  // r = 1: logits = u_hat . v0
  caps_route<<<grid, 128, 0, stream>>>(x, Wm, vsum, s_part, 1);
  caps_squash<<<Bsz * Ccap, 32, 0, stream>>>(s_part, bias, vsum, out, 0); // vsum = v0+v1
  // r = 2: logits = u_hat . (v0+v1)
  caps_route<<<grid, 128, 0, stream>>>(x, Wm, vsum, s_part, 1);
  caps_squash<<<Bsz * Ccap, 32, 0, stream>>>(s_part, bias, vsum, out, 1); // write out
}